// MultiHeadCrossAttention_10204842295866
// MI455X (gfx1250) — compile-verified
//
#include <hip/hip_runtime.h>

// MultiHeadCrossAttention (no softmax) == linear attention:
//   out[b,:,h,:] = X_h @ N[b,h],
//   N[b,h] = Wq^T (0.125 * Wk (C_h^T C_h) Wv^T)
// Bandwidth-bound (~50MB @ 23.3TB/s); fp32 WMMA 16x16x4 for the two big passes.

#define B_   2
#define SQ_  2048
#define SKV_ 2048
#define H_   16
#define DH_  64
#define D_   (H_ * DH_)   // 1024
#define KCHUNKS_ 8        // k-split of SKV for the Gram pass

typedef __attribute__((ext_vector_type(2))) float v2f;
typedef __attribute__((ext_vector_type(8))) float v8f;

static __device__ __forceinline__ v8f wmma_f32(v2f a, v2f b, v8f c) {
    // D(16x16 f32) = A(16x4 f32) * B(4x16 f32) + C
    return __builtin_amdgcn_wmma_f32_16x16x4_f32(
        /*neg_a=*/false, a, /*neg_b=*/false, b,
        /*c_mod=*/(short)0, c, /*reuse_a=*/false, /*reuse_b=*/false);
}

// ---------------------------------------------------------------------------
// Kernel 1: partial Gram matrices  Gpart[bh][kc] = C[kc-rows]^T C[kc-rows]
// grid (32, 8), block 128 (4 waves). Each wave covers 64 k-rows, full 64x64 G.
// ---------------------------------------------------------------------------
__global__ __launch_bounds__(128) void g_partial_kernel(
    const float* __restrict__ ctx, float* __restrict__ Gws) {
    const int bh   = blockIdx.x;        // 0..31
    const int b    = bh >> 4;
    const int h    = bh & 15;
    const int kc   = blockIdx.y;        // 0..7 (256 rows each)
    const int wave = threadIdx.x >> 5;  // 0..3 (64 rows each)
    const int lane = threadIdx.x & 31;
    const int half = lane >> 4;         // 0 / 1
    const int m    = lane & 15;

    const float* Cb = ctx + ((size_t)b * SKV_ + (size_t)kc * 256 + (size_t)wave * 64) * D_
                          + (size_t)h * DH_;

    const v8f vzero = {0.f,0.f,0.f,0.f,0.f,0.f,0.f,0.f};
    v8f acc[4][4];
    #pragma unroll
    for (int i = 0; i < 4; ++i)
        #pragma unroll
        for (int j = 0; j < 4; ++j) acc[i][j] = vzero;

    // 16 k-steps of 4 rows
    for (int ks = 0; ks < 64; ks += 4) {
        const float* row0 = Cb + (size_t)(ks + 2 * half) * D_;
        v2f f[4];
        #pragma unroll
        for (int c = 0; c < 4; ++c) {
            // A-frag (transposed C) and B-frag share this gather:
            // lane(half,m) holds C[k + 2*half + {0,1}, c*16 + m]
            f[c].x = row0[c * 16 + m];
            f[c].y = row0[D_ + c * 16 + m];
        }
        #pragma unroll
        for (int i = 0; i < 4; ++i)
            #pragma unroll
            for (int j = 0; j < 4; ++j)
                acc[i][j] = wmma_f32(f[i], f[j], acc[i][j]);
    }

    // per-wave partials -> LDS, tree-sum, one deterministic store per block
    __shared__ float part[4][64 * 64];   // 64 KB
    #pragma unroll
    for (int i = 0; i < 4; ++i)
        #pragma unroll
        for (int j = 0; j < 4; ++j)
            #pragma unroll
            for (int r = 0; r < 8; ++r)
                part[wave][(i * 16 + r + 8 * half) * 64 + j * 16 + m] = acc[i][j][r];
    __syncthreads();

    float* Gd = Gws + ((size_t)bh * KCHUNKS_ + kc) * 4096;
    for (int idx = threadIdx.x; idx < 4096; idx += 128)
        Gd[idx] = part[0][idx] + part[1][idx] + part[2][idx] + part[3][idx];
}

// ---------------------------------------------------------------------------
// Kernel 2: N[b,h] = Wq^T * (0.125 * Wk * G * Wv^T).  32 blocks x 256 threads.
// Tiny (50 MFLOP total) — plain FMAs through LDS.
// ---------------------------------------------------------------------------
__global__ __launch_bounds__(256) void n_kernel(
    const float* __restrict__ Gws,
    const float* __restrict__ Wq, const float* __restrict__ Wk,
    const float* __restrict__ Wv, float* __restrict__ Nws) {
    const int bh = blockIdx.x;
    const int h  = bh & 15;
    __shared__ float sA[4096];
    __shared__ float sB[4096];

    // reduce the 8 k-chunk partials of G
    const float* Gd = Gws + (size_t)bh * KCHUNKS_ * 4096;
    for (int i = threadIdx.x; i < 4096; i += 256) {
        float s = 0.f;
        #pragma unroll
        for (int c = 0; c < KCHUNKS_; ++c) s += Gd[(size_t)c * 4096 + i];
        sA[i] = s;
    }
    __syncthreads();

    // T1 = G * Wv^T : T1[d1,f] = sum_d2 G[d1,d2] * Wv[h][f][d2]
    const float* wv = Wv + (size_t)h * 4096;
    for (int i = threadIdx.x; i < 4096; i += 256) {
        const int d1 = i >> 6, f = i & 63;
        float s = 0.f;
        for (int d2 = 0; d2 < 64; ++d2) s += sA[d1 * 64 + d2] * wv[f * 64 + d2];
        sB[i] = s;
    }
    __syncthreads();

    // T2 = Wk * T1 : T2[e,f] = sum_d1 Wk[h][e][d1] * T1[d1,f]
    const float* wk = Wk + (size_t)h * 4096;
    for (int i = threadIdx.x; i < 4096; i += 256) {
        const int e = i >> 6, f = i & 63;
        float s = 0.f;
        for (int d1 = 0; d1 < 64; ++d1) s += wk[e * 64 + d1] * sB[d1 * 64 + f];
        sA[i] = s;                       // sA no longer needed as G
    }
    __syncthreads();

    // N = 0.125 * Wq^T * T2 : N[d,f] = 0.125 * sum_e Wq[h][e][d] * T2[e,f]
    const float* wq = Wq + (size_t)h * 4096;
    float* Nd = Nws + (size_t)bh * 4096;
    for (int i = threadIdx.x; i < 4096; i += 256) {
        const int d = i >> 6, f = i & 63;
        float s = 0.f;
        for (int e = 0; e < 64; ++e) s += wq[e * 64 + d] * sA[e * 64 + f];
        Nd[i] = 0.125f * s;
    }
}

// ---------------------------------------------------------------------------
// Kernel 3: out[b, q, h, :] = X[b, q, h-slice] @ N[b,h]
// grid (32, 32), block 128 (4 waves x 16 q-rows = 64 q-rows per block).
// ---------------------------------------------------------------------------
__global__ __launch_bounds__(128) void out_kernel(
    const float* __restrict__ X, const float* __restrict__ Nws,
    float* __restrict__ out) {
    const int bh   = blockIdx.x;
    const int b    = bh >> 4;
    const int h    = bh & 15;
    const int wave = threadIdx.x >> 5;
    const int lane = threadIdx.x & 31;
    const int half = lane >> 4;
    const int m    = lane & 15;

    __shared__ float sN[4096];           // 16 KB
    const float* Nd = Nws + (size_t)bh * 4096;
    for (int i = threadIdx.x; i < 4096; i += 128) sN[i] = Nd[i];
    __syncthreads();

    const int q0 = blockIdx.y * 64 + wave * 16;
    const float* Xr = X + ((size_t)b * SQ_ + q0 + m) * D_ + (size_t)h * DH_;

    const v8f vzero = {0.f,0.f,0.f,0.f,0.f,0.f,0.f,0.f};
    v8f acc[4];
    #pragma unroll
    for (int j = 0; j < 4; ++j) acc[j] = vzero;

    for (int k0 = 0; k0 < 64; k0 += 4) {
        // A-frag: contiguous pair X[q0+m, k0 + 2*half + {0,1}] -> one b64 load
        v2f a = *(const v2f*)(Xr + k0 + 2 * half);
        #pragma unroll
        for (int j = 0; j < 4; ++j) {
            v2f bf;
            bf.x = sN[(k0 + 2 * half) * 64 + j * 16 + m];
            bf.y = sN[(k0 + 2 * half + 1) * 64 + j * 16 + m];
            acc[j] = wmma_f32(a, bf, acc[j]);
        }
    }

    float* Or = out + ((size_t)b * SQ_ + q0 + 8 * half) * D_ + (size_t)h * DH_;
    #pragma unroll
    for (int j = 0; j < 4; ++j)
        #pragma unroll
        for (int r = 0; r < 8; ++r)
            Or[(size_t)r * D_ + j * 16 + m] = acc[j][r];
}

// ---------------------------------------------------------------------------
extern "C" void kernel_launch(void* const* d_in, const int* in_sizes, int n_in,
                              void* d_out, int out_size, void* d_ws, size_t ws_size,
                              hipStream_t stream) {
    (void)in_sizes; (void)n_in; (void)out_size; (void)ws_size;
    const float* ctx = (const float*)d_in[0];
    const float* X   = (const float*)d_in[1];
    const float* Wq  = (const float*)d_in[2];
    const float* Wk  = (const float*)d_in[3];
    const float* Wv  = (const float*)d_in[4];
    float* out = (float*)d_out;

    // scratch layout: [32*8*4096] G partials (4MB) | [32*4096] N (512KB)
    float* Gws = (float*)d_ws;
    float* Nws = Gws + (size_t)32 * KCHUNKS_ * 4096;

    g_partial_kernel<<<dim3(32, KCHUNKS_), 128, 0, stream>>>(ctx, Gws);
    n_kernel<<<dim3(32), 256, 0, stream>>>(Gws, Wq, Wk, Wv, Nws);
    out_kernel<<<dim3(32, 32), 128, 0, stream>>>(X, Nws, out);
}